// CausalDepthwiseConv_54657753809306
// MI455X (gfx1250) — compile-verified
//
#include <hip/hip_runtime.h>
#include <stdint.h>

// ---------------------------------------------------------------------------
// Causal depthwise conv (K=4) over seq:  out[b,s,h,d] = sum_k w[k,h,d]*x[b,s-k,h,d]
// Pure streaming / bandwidth-bound kernel for MI455X (gfx1250, wave32).
// Data path: async global->LDS (ASYNCcnt) double-buffered pipeline, B128 ops,
// register sliding window along seq, non-temporal B128 stores.
// ---------------------------------------------------------------------------

typedef float v4f __attribute__((ext_vector_type(4)));
typedef int   v4i __attribute__((vector_size(16)));   // pointee type the builtin wants

#define BATCH   4
#define SEQ     4096
#define C4      512      // (16 heads * 128 d_head) / 4 floats = float4 groups per row
#define BLOCK   256      // threads per block (8 wave32)
#define CHUNK   64       // seq rows per block
#define G       8        // rows per pipeline stage
#define NST     (CHUNK / G)

#if defined(__has_builtin)
#  if __has_builtin(__builtin_amdgcn_global_load_async_to_lds_b128)
#    define HAVE_ASYNC 1
#  else
#    define HAVE_ASYNC 0
#  endif
#else
#  define HAVE_ASYNC 0
#endif

#if HAVE_ASYNC
#  if __has_builtin(__builtin_amdgcn_s_wait_asynccnt)
#    define WAIT_ASYNC(n) __builtin_amdgcn_s_wait_asynccnt(n)
#  else
#    define WAIT_ASYNC(n) asm volatile("s_wait_asynccnt %0" ::"n"(n) : "memory")
#  endif
#else
#  define WAIT_ASYNC(n) do {} while (0)
#endif

#if defined(__has_builtin) && __has_builtin(__builtin_amdgcn_s_wait_dscnt)
#  define WAIT_DS0() __builtin_amdgcn_s_wait_dscnt(0)
#else
#  define WAIT_DS0() asm volatile("s_wait_dscnt 0" ::: "memory")
#endif

__device__ __forceinline__ void async_copy16(const v4f* src, v4f* dst) {
#if HAVE_ASYNC
  // GLOBAL_LOAD_ASYNC_TO_LDS_B128: each lane copies 16B, tracked on ASYNCcnt.
  __builtin_amdgcn_global_load_async_to_lds_b128(
      (__attribute__((address_space(1))) v4i*)src,
      (__attribute__((address_space(3))) v4i*)dst,
      /*offset=*/0, /*cpol=*/0);
#else
  *dst = *src;  // synchronous fallback (still correct, per-thread slot)
#endif
}

__global__ __launch_bounds__(BLOCK)
void causal_dwconv_k(const v4f* __restrict__ x4,
                     const v4f* __restrict__ w4,
                     v4f* __restrict__ o4) {
  // double-buffered staging: 2 stages * G rows * 256 float4 = 64 KB LDS
  __shared__ v4f lds[2 * G * BLOCK];

  const int tid = threadIdx.x;
  const int cb  = blockIdx.x * BLOCK + tid;          // channel float4 group [0, C4)
  const int b   = blockIdx.z;
  const int s0  = blockIdx.y * CHUNK;
  const size_t rowbase = (size_t)b * SEQ * C4 + cb;  // + s*C4 selects the row

  // per-channel taps (32 KB total, cache resident)
  const v4f w0 = w4[0 * C4 + cb];
  const v4f w1 = w4[1 * C4 + cb];
  const v4f w2 = w4[2 * C4 + cb];
  const v4f w3 = w4[3 * C4 + cb];

  // causal halo: rows s0-1..s0-3 (zero below 0); uniform branches per block
  v4f xm1 = {0.f, 0.f, 0.f, 0.f};
  v4f xm2 = {0.f, 0.f, 0.f, 0.f};
  v4f xm3 = {0.f, 0.f, 0.f, 0.f};
  if (s0 >= 1) xm1 = x4[rowbase + (size_t)(s0 - 1) * C4];
  if (s0 >= 2) xm2 = x4[rowbase + (size_t)(s0 - 2) * C4];
  if (s0 >= 3) xm3 = x4[rowbase + (size_t)(s0 - 3) * C4];

  // prime the pipeline: stage 0 -> buffer 0
#pragma unroll
  for (int r = 0; r < G; ++r)
    async_copy16(&x4[rowbase + (size_t)(s0 + r) * C4], &lds[r * BLOCK + tid]);

  for (int st = 0; st < NST; ++st) {
    const int cur = st & 1;
    if (st + 1 < NST) {
      const int nxt = (st + 1) & 1;
#pragma unroll
      for (int r = 0; r < G; ++r)
        async_copy16(&x4[rowbase + (size_t)(s0 + (st + 1) * G + r) * C4],
                     &lds[(nxt * G + r) * BLOCK + tid]);
      WAIT_ASYNC(G);   // async loads complete in order -> stage `st` is resident
    } else {
      WAIT_ASYNC(0);
    }

#pragma unroll
    for (int r = 0; r < G; ++r) {
      const v4f x0 = lds[(cur * G + r) * BLOCK + tid];
      const v4f o  = w0 * x0 + w1 * xm1 + w2 * xm2 + w3 * xm3;
      xm3 = xm2; xm2 = xm1; xm1 = x0;
      __builtin_nontemporal_store(
          o, &o4[rowbase + (size_t)(s0 + st * G + r) * C4]);
    }
    // WAR guard: our ds reads must land before stage st+2 overwrites this buffer
    WAIT_DS0();
  }
}

extern "C" void kernel_launch(void* const* d_in, const int* in_sizes, int n_in,
                              void* d_out, int out_size, void* d_ws, size_t ws_size,
                              hipStream_t stream) {
  (void)in_sizes; (void)n_in; (void)out_size; (void)d_ws; (void)ws_size;
  const v4f* x4 = (const v4f*)d_in[0];   // x: (4,4096,16,128) f32
  const v4f* w4 = (const v4f*)d_in[1];   // weight: (4,16,128) f32
  v4f* o4 = (v4f*)d_out;

  dim3 grid(C4 / BLOCK, SEQ / CHUNK, BATCH);  // (2, 64, 4) = 512 blocks
  causal_dwconv_k<<<grid, dim3(BLOCK), 0, stream>>>(x4, w4, o4);
}